// GCN_Encoder_73598559584319
// MI455X (gfx1250) — compile-verified
//
#include <hip/hip_runtime.h>
#include <math.h>

typedef __attribute__((ext_vector_type(2))) float v2f;
typedef __attribute__((ext_vector_type(8))) float v8f;

// ---------------------------------------------------------------------------
// GEMM: C[N,M] = A[N,K] @ W[K,M] + bias   (fp32, V_WMMA_F32_16X16X4_F32)
// One wave32 per 16x16 output tile. A-frag: 16x4 (2 VGPR/lane), B-frag: 4x16.
// Layouts per CDNA5 ISA 7.12.2:
//   A 16x4:  lanes 0-15 -> M=lane, VGPR0/1 = K0/K1 ; lanes 16-31 -> K2/K3
//   B 4x16:  lanes 0-15 -> N=lane, VGPR0/1 = K0/K1 ; lanes 16-31 -> K2/K3
//   C/D:     VGPR r -> M = r + 8*(lane>=16), N = lane%16
// ---------------------------------------------------------------------------
__global__ void gemm_bias_wmma(const float* __restrict__ A,
                               const float* __restrict__ W,
                               const float* __restrict__ bias,
                               float* __restrict__ C,
                               int N, int K, int M) {
  const int waveId = (int)((blockIdx.x * blockDim.x + threadIdx.x) >> 5);
  const int lane   = threadIdx.x & 31;
  const int tilesM = M >> 4;
  const int tileR  = waveId / tilesM;
  const int tileC  = waveId - tileR * tilesM;
  if (tileR * 16 >= N) return;             // wave-uniform: EXEC stays all-ones
  const int row0 = tileR << 4;
  const int col0 = tileC << 4;
  const int half = lane >> 4;              // 0: lanes 0-15, 1: lanes 16-31
  const int l16  = lane & 15;

  v8f acc = {};
  const float* arow = A + (size_t)(row0 + l16) * K;
  for (int kb = 0; kb < K; kb += 4) {
    const int ka = kb + (half << 1);
    v2f a = *(const v2f*)(arow + ka);                  // 8B-aligned (K%4==0)
    v2f b;
    b.x = W[(size_t)ka * M + col0 + l16];
    b.y = W[(size_t)(ka + 1) * M + col0 + l16];
    acc = __builtin_amdgcn_wmma_f32_16x16x4_f32(false, a, false, b,
                                                (short)0, acc, false, false);
  }
  const int colw  = col0 + l16;
  const int rbase = row0 + (half << 3);
  const float bc  = bias[colw];
#pragma unroll
  for (int r = 0; r < 8; ++r)
    C[(size_t)(rbase + r) * M + colw] = acc[r] + bc;
}

// ---------------------------------------------------------------------------
// Order-preserving float <-> uint mapping for atomic segment-max.
// ---------------------------------------------------------------------------
__device__ __forceinline__ unsigned fmap(float f) {
  unsigned u = __float_as_uint(f);
  return (u & 0x80000000u) ? ~u : (u | 0x80000000u);
}
__device__ __forceinline__ float funmap(unsigned u) {
  return __uint_as_float((u & 0x80000000u) ? (u & 0x7FFFFFFFu) : ~u);
}

// score[e] = dot(q[dst], k[src]) / sqrt(d).  One wave32 per edge.
__global__ void edge_scores(const float* __restrict__ q,
                            const float* __restrict__ k,
                            const int* __restrict__ src,
                            const int* __restrict__ dst,
                            float* __restrict__ score,
                            int E, int N, int d, float scale) {
  const int wid  = (int)((blockIdx.x * blockDim.x + threadIdx.x) >> 5);
  const int lane = threadIdx.x & 31;
  const int total = E + N;
  if (wid >= total) return;
  int s, t;
  if (wid < E) { s = src[wid]; t = dst[wid]; } else { s = t = wid - E; }
  const float* qr = q + (size_t)t * d;
  const float* kr = k + (size_t)s * d;
  float acc = 0.f;
  for (int c = lane; c < d; c += 32) acc += qr[c] * kr[c];
#pragma unroll
  for (int off = 16; off > 0; off >>= 1) acc += __shfl_xor(acc, off, 32);
  if (lane == 0) score[wid] = acc * scale;
}

__global__ void init_seg(unsigned* __restrict__ m, float* __restrict__ den, int N) {
  int i = blockIdx.x * blockDim.x + threadIdx.x;
  if (i < N) { m[i] = 0u; den[i] = 0.f; }
}

__global__ void seg_max(const float* __restrict__ score, const int* __restrict__ dst,
                        unsigned* __restrict__ m, int E, int N) {
  int e = blockIdx.x * blockDim.x + threadIdx.x;
  if (e >= E + N) return;
  int t = (e < E) ? dst[e] : (e - E);
  atomicMax(&m[t], fmap(score[e]));
}

__global__ void seg_expsum(float* __restrict__ score, const int* __restrict__ dst,
                           const unsigned* __restrict__ m, float* __restrict__ den,
                           int E, int N) {
  int e = blockIdx.x * blockDim.x + threadIdx.x;
  if (e >= E + N) return;
  int t = (e < E) ? dst[e] : (e - E);
  float ex = expf(score[e] - funmap(m[t]));
  score[e] = ex;
  atomicAdd(&den[t], ex);
}

__global__ void alpha_norm(float* __restrict__ exv, const float* __restrict__ den,
                           const int* __restrict__ dst, int E, int N) {
  int e = blockIdx.x * blockDim.x + threadIdx.x;
  if (e >= E + N) return;
  int t = (e < E) ? dst[e] : (e - E);
  exv[e] = exv[e] / den[t];
}

// out[dst] += alpha[e] * v[src]  (out pre-initialized with skip projection)
__global__ void scatter_av(const float* __restrict__ alpha, const float* __restrict__ v,
                           const int* __restrict__ src, const int* __restrict__ dst,
                           float* __restrict__ out, int E, int N, int d, int logd) {
  long long idx = (long long)blockIdx.x * blockDim.x + threadIdx.x;
  long long total = ((long long)(E + N)) << logd;
  if (idx >= total) return;
  int e = (int)(idx >> logd);
  int c = (int)(idx & (d - 1));
  int s, t;
  if (e < E) { s = src[e]; t = dst[e]; } else { s = t = e - E; }
  atomicAdd(&out[(size_t)t * d + c], alpha[e] * v[(size_t)s * d + c]);
}

// ---------------------------------------------------------------------------
// BatchNorm (biased variance, matching jnp.var ddof=0) + ReLU, in place.
// ---------------------------------------------------------------------------
__global__ void zero_f(float* p, int n) {
  int i = blockIdx.x * blockDim.x + threadIdx.x;
  if (i < n) p[i] = 0.f;
}

__global__ void bn_stats(const float* __restrict__ x, float* __restrict__ sum,
                         float* __restrict__ sumsq, int N, int d) {
  int c = threadIdx.x;                      // blockDim.x == d (64 or 128)
  float s = 0.f, s2 = 0.f;
  for (int r = blockIdx.x; r < N; r += gridDim.x) {
    float v = x[(size_t)r * d + c];
    s += v; s2 += v * v;
  }
  atomicAdd(&sum[c], s);
  atomicAdd(&sumsq[c], s2);
}

__global__ void bn_finalize(float* __restrict__ sums, const float* __restrict__ g,
                            const float* __restrict__ be, int N, int d) {
  int c = threadIdx.x;
  if (c >= d) return;
  float inv_n = 1.f / (float)N;
  float mean  = sums[c] * inv_n;
  float var   = sums[d + c] * inv_n - mean * mean;
  float sc    = g[c] * rsqrtf(var + 1e-5f);
  sums[c]     = sc;                // scale
  sums[d + c] = be[c] - mean * sc; // shift
}

__global__ void bn_apply_relu(float* __restrict__ x, const float* __restrict__ scale,
                              const float* __restrict__ shift, int total, int dmask) {
  int i = blockIdx.x * blockDim.x + threadIdx.x;
  if (i >= total) return;
  int c = i & dmask;
  float y = x[i] * scale[c] + shift[c];
  x[i] = y > 0.f ? y : 0.f;
}

// ---------------------------------------------------------------------------
// Host orchestration
// ---------------------------------------------------------------------------
static void launch_gemm(const float* A, const float* W, const float* b, float* C,
                        int N, int K, int M, hipStream_t st) {
  int tiles  = (N / 16) * (M / 16);
  int blocks = (tiles * 32 + 255) / 256;
  gemm_bias_wmma<<<blocks, 256, 0, st>>>(A, W, b, C, N, K, M);
}

static void tconv(const float* X, int din, int dout,
                  const float* Wq, const float* bq, const float* Wk, const float* bk,
                  const float* Wv, const float* bv, const float* Ws, const float* bs,
                  float* bufQ, float* bufK, float* bufV, float* out,
                  float* exv, unsigned* m, float* den,
                  const int* src, const int* dst, int E, int N, hipStream_t st) {
  launch_gemm(X, Wq, bq, bufQ, N, din, dout, st);
  launch_gemm(X, Wk, bk, bufK, N, din, dout, st);
  launch_gemm(X, Wv, bv, bufV, N, din, dout, st);
  launch_gemm(X, Ws, bs, out,  N, din, dout, st);   // skip path seeds the output

  const int total = E + N;
  {
    int blocks = (total * 32 + 255) / 256;      // wave per edge
    edge_scores<<<blocks, 256, 0, st>>>(bufQ, bufK, src, dst, exv, E, N, dout,
                                        1.f / sqrtf((float)dout));
  }
  init_seg<<<(N + 255) / 256, 256, 0, st>>>(m, den, N);
  seg_max<<<(total + 255) / 256, 256, 0, st>>>(exv, dst, m, E, N);
  seg_expsum<<<(total + 255) / 256, 256, 0, st>>>(exv, dst, m, den, E, N);
  alpha_norm<<<(total + 255) / 256, 256, 0, st>>>(exv, den, dst, E, N);
  const int logd = (dout == 128) ? 7 : 6;
  long long tot = ((long long)total) << logd;
  scatter_av<<<(int)((tot + 255) / 256), 256, 0, st>>>(exv, bufV, src, dst, out,
                                                       E, N, dout, logd);
}

static void bn_relu(float* Xio, const float* g, const float* be, int d, int N,
                    float* sums, hipStream_t st) {
  zero_f<<<1, 2 * d, 0, st>>>(sums, 2 * d);
  bn_stats<<<256, d, 0, st>>>(Xio, sums, sums + d, N, d);
  bn_finalize<<<1, d, 0, st>>>(sums, g, be, N, d);
  bn_apply_relu<<<(N * d + 255) / 256, 256, 0, st>>>(Xio, sums, sums + d, N * d, d - 1);
}

extern "C" void kernel_launch(void* const* d_in, const int* in_sizes, int n_in,
                              void* d_out, int out_size, void* d_ws, size_t ws_size,
                              hipStream_t stream) {
  (void)n_in; (void)out_size; (void)ws_size;
  const float* x  = (const float*)d_in[0];
  const int*   ei = (const int*)d_in[1];
  const int E = in_sizes[1] / 2;
  const int N = in_sizes[0] / 128;
  const int* src = ei;
  const int* dst = ei + E;

  const float* lin_W = (const float*)d_in[2];
  const float* lin_b = (const float*)d_in[3];

  const float *Wq[3], *Wk[3], *Wv[3], *Ws[3], *bq[3], *bk[3], *bv[3], *bs[3];
  for (int l = 0; l < 3; ++l) {
    int base = 4 + l * 8;
    Wq[l] = (const float*)d_in[base + 0];
    Wk[l] = (const float*)d_in[base + 1];
    Wv[l] = (const float*)d_in[base + 2];
    Ws[l] = (const float*)d_in[base + 3];
    bq[l] = (const float*)d_in[base + 4];
    bk[l] = (const float*)d_in[base + 5];
    bv[l] = (const float*)d_in[base + 6];
    bs[l] = (const float*)d_in[base + 7];
  }
  const float* g1  = (const float*)d_in[28];
  const float* be1 = (const float*)d_in[29];
  const float* g2  = (const float*)d_in[30];
  const float* be2 = (const float*)d_in[31];

  char* p = (char*)d_ws;
  auto alloc = [&](size_t bytes) -> char* {
    char* r = p; p += (bytes + 255) & ~(size_t)255; return r;
  };
  float*    bufA = (float*)alloc((size_t)N * 128 * 4);
  float*    bufB = (float*)alloc((size_t)N * 128 * 4);
  float*    bufQ = (float*)alloc((size_t)N * 128 * 4);
  float*    bufK = (float*)alloc((size_t)N * 128 * 4);
  float*    bufV = (float*)alloc((size_t)N * 128 * 4);
  float*    exv  = (float*)alloc((size_t)(E + N) * 4);
  unsigned* m    = (unsigned*)alloc((size_t)N * 4);
  float*    den  = (float*)alloc((size_t)N * 4);
  float*    sums = (float*)alloc(1024);

  // h = x @ lin_W + lin_b       [N,128] x [128,64]
  launch_gemm(x, lin_W, lin_b, bufA, N, 128, 64, stream);

  // Layer 1: 64 -> 128, BN+ReLU
  tconv(bufA, 64, 128, Wq[0], bq[0], Wk[0], bk[0], Wv[0], bv[0], Ws[0], bs[0],
        bufQ, bufK, bufV, bufB, exv, m, den, src, dst, E, N, stream);
  bn_relu(bufB, g1, be1, 128, N, sums, stream);

  // Layer 2: 128 -> 64, BN+ReLU
  tconv(bufB, 128, 64, Wq[1], bq[1], Wk[1], bk[1], Wv[1], bv[1], Ws[1], bs[1],
        bufQ, bufK, bufV, bufA, exv, m, den, src, dst, E, N, stream);
  bn_relu(bufA, g2, be2, 64, N, sums, stream);

  // Layer 3: 64 -> 64, straight into d_out
  tconv(bufA, 64, 64, Wq[2], bq[2], Wk[2], bk[2], Wv[2], bv[2], Ws[2], bs[2],
        bufQ, bufK, bufV, (float*)d_out, exv, m, den, src, dst, E, N, stream);
}